// ID_21079699489011
// MI455X (gfx1250) — compile-verified
//
#include <hip/hip_runtime.h>
#include <stdint.h>

// Serial RK4 scan: x_{t+1} = RK4(x_t, u_t). Latency-bound on a ~10-deep FMA
// chain per step * 200k steps. One wave32 owns the whole recurrence; input
// stream u is double-buffered into LDS via CDNA5 async global->LDS DMA
// (global_load_async_to_lds_b128 + s_wait_asynccnt) so no VMEM latency ever
// lands on the serial critical path. Compute + output store run under a
// single-lane EXEC mask per chunk.
//
// Depth-optimized RK4:
//   stage outputs computed directly as a = x + c*f(s) with pre-scaled coeffs
//   and off-chain constant terms ct2 = x2 + c*k4*Fs, ct3 = x3 + c*k6
//   (depend only on base state + Fs)  -> depth 3 per stage, no stage-add.
//   Final:  x' = (1/3)*(A + 2B + C - x) + (H/6)*K4
//   where the first term is assembled off-chain while K4=f(C) evaluates,
//   so the update costs depth 1 after K4. K4 comps 0/1 are C2,C3 (free).

namespace {
constexpr int   kChunk = 1024;          // floats per LDS buffer (4 KB)
constexpr float kH     = 0.1f;          // RK4 step
constexpr float kH2    = 0.05f;         // H/2
constexpr float kH6    = 0.1f / 6.0f;   // H/6
constexpr float kThird = 1.0f / 3.0f;
}

// One RK4 stage: a = xbase + c * f(s), pre-scaled coefficients.
// ck* = c*k*, cc34 = c*0.5*k3*k4, cc36 = c*k3*k6,
// ct2 = x2base + c*k4*Fs, ct3 = x3base + c*k6   (both off-chain).
// Critical depth 3 for comps 2/3, depth 1 for comps 0/1.
__device__ __forceinline__ void rk_stage(float s0, float s1, float s2, float s3,
                                         float x0, float x1,
                                         float ck1, float ck2, float ck5,
                                         float cc34, float cc36,
                                         float ct2, float ct3, float c,
                                         float& a0, float& a1, float& a2, float& a3) {
  a0 = fmaf(c, s2, x0);
  a1 = fmaf(c, s3, x1);
  a2 = fmaf(cc34, s1 * s1, fmaf(ck1, s2, fmaf(ck2, s0, ct2)));
  a3 = fmaf((1.0f - s0) * cc36, s1, fmaf(ck5, s3, ct3));
}

__global__ void __launch_bounds__(32, 1)
ID_21079699489011_rk4_scan(const float* __restrict__ u,
                           const float* __restrict__ k1p, const float* __restrict__ k2p,
                           const float* __restrict__ k3p, const float* __restrict__ k4p,
                           const float* __restrict__ k5p, const float* __restrict__ k6p,
                           float* __restrict__ out, int n) {
  __shared__ float sbuf[2][kChunk];
  const int lane = (int)threadIdx.x;

  const float k1 = *k1p, k2 = *k2p, k3 = *k3p;
  const float k4 = *k4p, k5 = *k5p, k6 = *k6p;
  const float c34 = 0.5f * k3 * k4;   // coeff of x1^2 in f2
  const float c36 = k3 * k6;          // coeff of (1-x0)*x1 in f3

  // Pre-scaled stage coefficients (hoisted, live in registers).
  const float h2k1 = kH2 * k1, h2k2 = kH2 * k2, h2k5 = kH2 * k5;
  const float h2c34 = kH2 * c34, h2c36 = kH2 * c36;
  const float h2k4 = kH2 * k4, h2k6 = kH2 * k6;
  const float hk1 = kH * k1, hk2 = kH * k2, hk5 = kH * k5;
  const float hc34 = kH * c34, hc36 = kH * c36;
  const float hk4 = kH * k4, hk6 = kH * k6;

  const uint64_t gbase  = (uint64_t)(uintptr_t)u;
  const uint64_t gclamp = gbase + (uint64_t)((n >= 4) ? (n - 4) : 0) * 4ull;

  // Async-copy kChunk floats of u into LDS buffer `buf`.
  // 32 lanes x 16B per global_load_async_to_lds_b128 = 512B/instr, 8 instrs/chunk.
  auto issue_chunk = [&](int buf, int elem_base) {
    const uint32_t lds0 = (uint32_t)(uintptr_t)(&sbuf[buf][0]);  // low 32b of flat LDS ptr == LDS byte addr
#pragma unroll
    for (int k = 0; k < kChunk / 128; ++k) {
      const int e = k * 128 + lane * 4;                   // 4 floats / lane, coalesced
      uint64_t g = gbase + (uint64_t)(elem_base + e) * 4ull;
      if (g > gclamp) g = gclamp;                         // tail chunk stays in-bounds
      const uint32_t l = lds0 + (uint32_t)(e * 4);
      asm volatile("global_load_async_to_lds_b128 %0, %1, off"
                   :: "v"(l), "v"(g) : "memory");
    }
  };

  float x0 = 0.f, x1 = 0.f, x2 = 0.f, x3 = 0.f;
  const int nchunk = (n + kChunk - 1) / kChunk;

  issue_chunk(0, 0);
  for (int c = 0; c < nchunk; ++c) {
    if (c + 1 < nchunk) {
      issue_chunk((c + 1) & 1, (c + 1) * kChunk);
      // loads retire in order: <=8 outstanding means chunk c fully in LDS
      asm volatile("s_wait_asynccnt 0x8" ::: "memory");
    } else {
      asm volatile("s_wait_asynccnt 0x0" ::: "memory");
    }

    const int rem = n - c * kChunk;
    const int m   = rem < kChunk ? rem : kChunk;
    const int cb  = c & 1;
    float4* __restrict__ orow = reinterpret_cast<float4*>(out) + c * kChunk;

    // Single divergence per chunk: whole compute loop runs with EXEC = lane0.
    if (lane == 0) {
#pragma unroll 2
      for (int i = 0; i < m; ++i) {
        const float Fs = sbuf[cb][i];     // broadcast ds_load_b32, off the chain

        // Off-chain per-step constants (depend only on base state + Fs).
        const float ct2a = fmaf(h2k4, Fs, x2);  // stages 1,2 (c = H/2)
        const float ct3a = x3 + h2k6;
        const float ct2c = fmaf(hk4, Fs, x2);   // stage 3   (c = H)
        const float ct3c = x3 + hk6;
        const float k4Fs = k4 * Fs;             // stage 4 (raw K4)

        float A0, A1, A2, A3, B0, B1, B2, B3, C0, C1, C2, C3;

        rk_stage(x0, x1, x2, x3, x0, x1, h2k1, h2k2, h2k5, h2c34, h2c36,
                 ct2a, ct3a, kH2, A0, A1, A2, A3);
        rk_stage(A0, A1, A2, A3, x0, x1, h2k1, h2k2, h2k5, h2c34, h2c36,
                 ct2a, ct3a, kH2, B0, B1, B2, B3);
        rk_stage(B0, B1, B2, B3, x0, x1, hk1, hk2, hk5, hc34, hc36,
                 ct2c, ct3c, kH, C0, C1, C2, C3);

        // K4 = f(C): comps 0/1 are C2,C3; comps 2/3 depth 3.
        const float K42 = fmaf(c34, C1 * C1, fmaf(k1, C2, fmaf(k2, C0, k4Fs)));
        const float K43 = fmaf((1.0f - C0) * c36, C1, fmaf(k5, C3, k6));

        // x' = (1/3)*(A + 2B + C - x) + (H/6)*K4  (first term off-chain w.r.t. K4)
        const float t0 = (fmaf(2.0f, B0, A0) - x0) + C0;
        const float t1 = (fmaf(2.0f, B1, A1) - x1) + C1;
        const float t2 = (fmaf(2.0f, B2, A2) - x2) + C2;
        const float t3 = (fmaf(2.0f, B3, A3) - x3) + C3;

        x0 = fmaf(kH6, C2, kThird * t0);
        x1 = fmaf(kH6, C3, kThird * t1);
        x2 = fmaf(kH6, K42, kThird * t2);
        x3 = fmaf(kH6, K43, kThird * t3);

        float4 v; v.x = x0; v.y = x1; v.z = x2; v.w = x3;
        orow[i] = v;                      // fire-and-forget b128 store
      }
    }
  }
}

extern "C" void kernel_launch(void* const* d_in, const int* in_sizes, int n_in,
                              void* d_out, int out_size, void* d_ws, size_t ws_size,
                              hipStream_t stream) {
  (void)n_in; (void)out_size; (void)d_ws; (void)ws_size;
  const float* u  = (const float*)d_in[0];
  const float* k1 = (const float*)d_in[1];
  const float* k2 = (const float*)d_in[2];
  const float* k3 = (const float*)d_in[3];
  const float* k4 = (const float*)d_in[4];
  const float* k5 = (const float*)d_in[5];
  const float* k6 = (const float*)d_in[6];
  float* out = (float*)d_out;
  const int n = in_sizes[0];

  // Inherently serial recurrence: exactly one wave32 owns it.
  ID_21079699489011_rk4_scan<<<1, 32, 0, stream>>>(u, k1, k2, k3, k4, k5, k6, out, n);
}